// HypProxyAnchor_62019327754785
// MI455X (gfx1250) — compile-verified
//
#include <hip/hip_runtime.h>
#include <math.h>

typedef __attribute__((ext_vector_type(2))) float v2f;
typedef __attribute__((ext_vector_type(8))) float v8f;

#define N_PTS 8192
#define N_CLS 8192
#define D_EMB 128
#define TM 64
#define TN 32
#define LDSP (D_EMB + 4)

__device__ __forceinline__ float waveReduceSum(float v) {
#pragma unroll
  for (int m = 16; m >= 1; m >>= 1) v += __shfl_xor(v, m, 32);
  return v;
}

__global__ void init_kernel(int* __restrict__ flags, float* __restrict__ accum) {
  int i = blockIdx.x * blockDim.x + threadIdx.x;
  if (i < N_CLS) flags[i] = 0;
  if (i < 2) accum[i] = 0.0f;
}

// One wave per proxy row: feature-clip -> expmap0 -> project into ball.
__global__ void proj_kernel(const float* __restrict__ Ptan, float* __restrict__ P,
                            float* __restrict__ y2) {
  const int lane = threadIdx.x & 31;
  const int wave = threadIdx.x >> 5;
  const int row  = blockIdx.x * 8 + wave;
  const float4 p4 = ((const float4*)(Ptan + (size_t)row * D_EMB))[lane];
  float ss = p4.x * p4.x + p4.y * p4.y + p4.z * p4.z + p4.w * p4.w;
  ss = waveReduceSum(ss);
  const float sqrt_c = 0.31622776601683794f;
  float nrm = fmaxf(sqrtf(ss), 1e-15f);
  float s1  = fminf(1.0f, 2.3f / nrm);          // feature clip
  float un  = fmaxf(nrm * s1, 1e-15f);
  float t   = tanhf(sqrt_c * un);
  float s2  = s1 * (t / (sqrt_c * un));         // combined scale tan->ball
  float gn  = fmaxf(t / sqrt_c, 1e-15f);        // norm of g
  float maxn = (1.0f - 1e-3f) / sqrt_c;
  float s3  = (gn > maxn) ? (maxn / gn) : 1.0f; // strict-interior projection
  float s   = s2 * s3;
  float4 g = make_float4(p4.x * s, p4.y * s, p4.z * s, p4.w * s);
  ((float4*)(P + (size_t)row * D_EMB))[lane] = g;
  if (lane == 0) { float gf = gn * s3; y2[row] = gf * gf; }
}

// One wave per X row: squared norms.
__global__ void rowsq_kernel(const float* __restrict__ X, float* __restrict__ x2) {
  const int lane = threadIdx.x & 31;
  const int wave = threadIdx.x >> 5;
  const int row  = blockIdx.x * 8 + wave;
  const float4 p4 = ((const float4*)(X + (size_t)row * D_EMB))[lane];
  float ss = p4.x * p4.x + p4.y * p4.y + p4.z * p4.z + p4.w * p4.w;
  ss = waveReduceSum(ss);
  if (lane == 0) x2[row] = ss;
}

__global__ void flag_kernel(const int* __restrict__ T, int* __restrict__ flags) {
  int i = blockIdx.x * blockDim.x + threadIdx.x;
  if (i < N_PTS) flags[T[i]] = 1;  // benign write race, all write 1
}

// 64x32 tile of the 8192x8192 distance matrix per block; f32 WMMA GEMM
// (X @ P^T) with fused Poincare-distance + softplus-loss epilogue.
__global__ __launch_bounds__(256) void gemm_loss_kernel(
    const float* __restrict__ X, const float* __restrict__ P,
    const float* __restrict__ x2, const float* __restrict__ y2,
    const int* __restrict__ T, float* __restrict__ accum) {
  __shared__ float Xs[TM * LDSP];
  __shared__ float Ps[TN * LDSP];
  __shared__ float sPos[256];
  __shared__ float sNeg[256];

  const int tid = threadIdx.x;
  const int rowBase = blockIdx.y * TM;
  const int colBase = blockIdx.x * TN;

  // Stage tiles into LDS (float4 lanes; rows padded to LDSP to stagger banks)
  for (int idx = tid; idx < TM * (D_EMB / 4); idx += 256) {
    int r = idx >> 5, c = idx & 31;
    float4 v = ((const float4*)(X + (size_t)(rowBase + r) * D_EMB))[c];
    *((float4*)&Xs[r * LDSP + c * 4]) = v;
  }
  for (int idx = tid; idx < TN * (D_EMB / 4); idx += 256) {
    int r = idx >> 5, c = idx & 31;
    float4 v = ((const float4*)(P + (size_t)(colBase + r) * D_EMB))[c];
    *((float4*)&Ps[r * LDSP + c * 4]) = v;
  }
  __syncthreads();

  const int lane = tid & 31;
  const int wave = tid >> 5;
  const int wm = wave & 3;   // 4 wave-rows of 16
  const int wn = wave >> 2;  // 2 wave-cols of 16
  const int half = lane >> 4;
  const int lid  = lane & 15;
  const int koff = half * 2;

  const float* xrow = &Xs[(wm * 16 + lid) * LDSP];  // A: M = lid
  const float* prow = &Ps[(wn * 16 + lid) * LDSP];  // B: N = lid

  v8f acc = {0.f, 0.f, 0.f, 0.f, 0.f, 0.f, 0.f, 0.f};
#pragma unroll
  for (int k0 = 0; k0 < D_EMB; k0 += 4) {
    v2f a = *((const v2f*)&xrow[k0 + koff]);  // K = k0+koff, k0+koff+1
    v2f b = *((const v2f*)&prow[k0 + koff]);
    acc = __builtin_amdgcn_wmma_f32_16x16x4_f32(false, a, false, b,
                                                (short)0, acc, false, false);
  }

  // Fused epilogue: xy -> Poincare distance -> softplus terms
  const float cc = 0.1f;
  const float sqrt_c = 0.31622776601683794f;
  const float inv_sqrt_c = 3.1622776601683795f;
  const int j = colBase + wn * 16 + lid;
  const float yj = y2[j];
  float posAcc = 0.0f, negAcc = 0.0f;
#pragma unroll
  for (int v = 0; v < 8; ++v) {
    const int i = rowBase + wm * 16 + v + half * 8;  // D layout: M = v (+8 hi half)
    const float xi = x2[i];
    const float xy = acc[v];
    const float Am = 1.0f - 2.0f * cc * xy + cc * yj;
    const float Bm = 1.0f - cc * xi;
    const float num2 = Am * Am * xi - 2.0f * Am * Bm * xy + Bm * Bm * yj;
    const float den  = 1.0f - 2.0f * cc * xy + cc * cc * xi * yj;
    const float ratio = sqrtf(fmaxf(num2, 1e-15f)) / den;
    float arg = sqrt_c * ratio;
    arg = fminf(fmaxf(arg, -1.0f + 1e-5f), 1.0f - 1e-5f);
    const float dist = inv_sqrt_c * logf((1.0f + arg) / (1.0f - arg)); // 2/sqrt_c*atanh
    const float spn = log1pf(expf(-dist));       // softplus(-d), d >= 0
    if (T[i] == j) posAcc += dist + spn;         // softplus(d)
    else           negAcc += spn;
  }

  sPos[tid] = posAcc; sNeg[tid] = negAcc;
  __syncthreads();
  for (int s = 128; s > 0; s >>= 1) {
    if (tid < s) { sPos[tid] += sPos[tid + s]; sNeg[tid] += sNeg[tid + s]; }
    __syncthreads();
  }
  if (tid == 0) {
    atomicAdd(&accum[0], sPos[0]);
    atomicAdd(&accum[1], sNeg[0]);
  }
}

__global__ void finalize_kernel(const int* __restrict__ flags,
                                const float* __restrict__ accum,
                                float* __restrict__ out) {
  __shared__ int sc[256];
  int cnt = 0;
  for (int i = threadIdx.x; i < N_CLS; i += 256) cnt += flags[i];
  sc[threadIdx.x] = cnt;
  __syncthreads();
  for (int s = 128; s > 0; s >>= 1) {
    if (threadIdx.x < s) sc[threadIdx.x] += sc[threadIdx.x + s];
    __syncthreads();
  }
  if (threadIdx.x == 0) {
    float nv = (float)sc[0];
    out[0] = accum[0] / nv + accum[1] / (float)N_CLS;
  }
}

extern "C" void kernel_launch(void* const* d_in, const int* in_sizes, int n_in,
                              void* d_out, int out_size, void* d_ws, size_t ws_size,
                              hipStream_t stream) {
  (void)in_sizes; (void)n_in; (void)out_size; (void)ws_size;
  const float* X    = (const float*)d_in[0];
  const int*   T    = (const int*)d_in[1];
  const float* Ptan = (const float*)d_in[2];
  float* out = (float*)d_out;

  // Workspace layout (bytes): P 4MB | x2 32KB | y2 32KB | flags 32KB | accum
  float* Pws   = (float*)d_ws;
  float* x2    = Pws + (size_t)N_CLS * D_EMB;
  float* y2    = x2 + N_PTS;
  int*   flags = (int*)(y2 + N_CLS);
  float* accum = (float*)(flags + N_CLS);

  init_kernel<<<(N_CLS + 255) / 256, 256, 0, stream>>>(flags, accum);
  proj_kernel<<<N_CLS / 8, 256, 0, stream>>>(Ptan, Pws, y2);
  rowsq_kernel<<<N_PTS / 8, 256, 0, stream>>>(X, x2);
  flag_kernel<<<N_PTS / 256, 256, 0, stream>>>(T, flags);

  dim3 grid(N_CLS / TN, N_PTS / TM);  // 256 x 128 blocks
  gemm_loss_kernel<<<grid, 256, 0, stream>>>(X, Pws, x2, y2, T, accum);

  finalize_kernel<<<1, 256, 0, stream>>>(flags, accum, out);
}